// TAWTrainer_54082228191620
// MI455X (gfx1250) — compile-verified
//
#include <hip/hip_runtime.h>

#define D_DIM 4096
#define KC 32
#define LDSP 36            // padded LDS row stride (floats): even + mult-of-4 -> 16B-aligned fills, conflict-free frags
#define EPSF 1e-16f

typedef __attribute__((ext_vector_type(2))) float v2f;
typedef __attribute__((ext_vector_type(8))) float v8f;
typedef int b128_t __attribute__((vector_size(16)));   // matches builtin's "int __vector(4)" pointee

#if __has_builtin(__builtin_amdgcn_global_load_async_to_lds_b128)
#define HAVE_ASYNC_LDS 1
#define AS1 __attribute__((address_space(1)))
#define AS3 __attribute__((address_space(3)))
#endif

#if defined(HAVE_ASYNC_LDS)
#if __has_builtin(__builtin_amdgcn_s_wait_asynccnt)
#define TAW_WAIT_ASYNC(n) __builtin_amdgcn_s_wait_asynccnt(n)   // n must be a literal
#else
#define TAW_WAIT_ASYNC(n) asm volatile("s_wait_asynccnt " #n ::: "memory")
#endif
#else
#define TAW_WAIT_ASYNC(n) ((void)0)
#endif

// ---------------- init: zero the dot accumulator matrix ----------------
__global__ __launch_bounds__(256)
void taw_init_kernel(float* __restrict__ dotacc, int total) {
    const int idx = blockIdx.x * 256 + threadIdx.x;
    if (idx < total) dotacc[idx] = 0.0f;
}

// ---------------- row squared norms for a and p ----------------
__global__ __launch_bounds__(256)
void taw_norms_kernel(const float* __restrict__ a, const float* __restrict__ p,
                      float* __restrict__ sqa, float* __restrict__ sqp,
                      int N, int D) {
    __shared__ float red[256];
    const int row = blockIdx.x;                  // 0..2N-1: first N rows of a, next N rows of p
    const int r   = (row < N) ? row : row - N;
    const float* base = ((row < N) ? a : p) + (size_t)r * D;
    float s = 0.0f;
    for (int k = threadIdx.x; k < D; k += 256) {
        float v = base[k];
        s += v * v;
    }
    red[threadIdx.x] = s;
    __syncthreads();
    for (int off = 128; off > 0; off >>= 1) {
        if (threadIdx.x < off) red[threadIdx.x] += red[threadIdx.x + off];
        __syncthreads();
    }
    if (threadIdx.x == 0) ((row < N) ? sqa : sqp)[r] = red[0];
}

// ---------------- fp32 WMMA GEMM, async-LDS double-buffered, K split across blockIdx.z ----------------
// Block: 128 threads = 4 waves, computes a 64x64 partial-dot tile over one K-half.
// Wave w: 32x32 register tile at (am = 32*(w>>1), cn = 32*(w&1)); 4 accumulators reuse a0/a1 x b0/b1.
__global__ __launch_bounds__(128)
void taw_gemm_kernel(const float* __restrict__ a, const float* __restrict__ p,
                     float* __restrict__ dotacc, int N, int D) {
    __shared__ __align__(16) float As[2][64][LDSP];
    __shared__ __align__(16) float Bs[2][64][LDSP];

    const int tid  = threadIdx.x;
    const int lane = tid & 31;
    const int w    = tid >> 5;                 // 0..3
    const int rowBase = blockIdx.y * 64;
    const int colBase = blockIdx.x * 64;
    const int kHalfLen = D >> 1;
    const int kBase    = blockIdx.z * kHalfLen;
    const int kSteps   = kHalfLen / KC;        // 64 panels

    const int am = (w >> 1) * 32;              // 0 or 32
    const int cn = (w & 1) * 32;               // 0 or 32
    const int m     = lane & 15;               // fragment row (A) / col (B)
    const int khalf = lane >> 4;               // half-wave K-pair select

    v8f c00 = {}, c01 = {}, c10 = {}, c11 = {};

    // Cooperative panel fill: 64x32 floats per panel = 512 float4; 4 per thread per panel.
    auto fill = [&](int buf, int kOff) {
#pragma unroll
        for (int l = 0; l < 4; ++l) {
            const int idx = tid + 128 * l;     // 0..511
            const int r   = idx >> 3;          // 0..63
            const int f   = (idx & 7) * 4;     // 0..28
            const float* ga = a + (size_t)(rowBase + r) * D + kOff + f;
            const float* gp = p + (size_t)(colBase + r) * D + kOff + f;
#if defined(HAVE_ASYNC_LDS)
            __builtin_amdgcn_global_load_async_to_lds_b128(
                (AS1 b128_t*)ga, (AS3 b128_t*)&As[buf][r][f], 0, 0);
            __builtin_amdgcn_global_load_async_to_lds_b128(
                (AS1 b128_t*)gp, (AS3 b128_t*)&Bs[buf][r][f], 0, 0);
#else
            *(float4*)&As[buf][r][f] = *(const float4*)ga;
            *(float4*)&Bs[buf][r][f] = *(const float4*)gp;
#endif
        }
    };

    fill(0, kBase);                            // prologue: 8 asyncs outstanding
    for (int i = 0; i < kSteps; ++i) {
        const int buf = i & 1;
        if (i + 1 < kSteps) {
            fill(buf ^ 1, kBase + (i + 1) * KC);   // overlap next panel with this compute
            TAW_WAIT_ASYNC(8);                     // current panel done; 8 newer asyncs in flight
        } else {
            TAW_WAIT_ASYNC(0);
        }
        __syncthreads();                           // panel visible to all 4 waves

#pragma unroll
        for (int kk = 0; kk < KC; kk += 4) {
            const v2f a0 = *(const v2f*)&As[buf][am +      m][kk + 2 * khalf];
            const v2f a1 = *(const v2f*)&As[buf][am + 16 + m][kk + 2 * khalf];
            const v2f b0 = *(const v2f*)&Bs[buf][cn +      m][kk + 2 * khalf];
            const v2f b1 = *(const v2f*)&Bs[buf][cn + 16 + m][kk + 2 * khalf];
            c00 = __builtin_amdgcn_wmma_f32_16x16x4_f32(false, a0, false, b0, (short)0, c00, false, false);
            c01 = __builtin_amdgcn_wmma_f32_16x16x4_f32(false, a0, false, b1, (short)0, c01, false, false);
            c10 = __builtin_amdgcn_wmma_f32_16x16x4_f32(false, a1, false, b0, (short)0, c10, false, false);
            c11 = __builtin_amdgcn_wmma_f32_16x16x4_f32(false, a1, false, b1, (short)0, c11, false, false);
        }
        __syncthreads();                           // all reads done before next fill overwrites buf
    }

    // Epilogue: combine the two K-slices. Exactly 2 commutative fp32 adds per element on a
    // zeroed buffer -> bitwise deterministic.
    const int gn = colBase + cn + m;
#pragma unroll
    for (int r = 0; r < 8; ++r) {
        const int gm0 = rowBase + am + r + 8 * khalf;
        const int gm1 = gm0 + 16;
        atomicAdd(&dotacc[(size_t)gm0 * N + gn],      c00[r]);
        atomicAdd(&dotacc[(size_t)gm0 * N + gn + 16], c01[r]);
        atomicAdd(&dotacc[(size_t)gm1 * N + gn],      c10[r]);
        atomicAdd(&dotacc[(size_t)gm1 * N + gn + 16], c11[r]);
    }
}

// ---------------- dot -> distance (in place) ----------------
__global__ __launch_bounds__(256)
void taw_dist_kernel(float* __restrict__ dotdist,
                     const float* __restrict__ sqa, const float* __restrict__ sqp, int N) {
    const int idx = blockIdx.x * 256 + threadIdx.x;
    if (idx < N * N) {
        const int i = idx / N;
        const int j = idx - i * N;
        float d2 = sqa[i] + sqp[j] - 2.0f * dotdist[idx];
        d2 = fmaxf(d2, 0.0f);
        dotdist[idx] = (d2 == 0.0f) ? 0.0f : sqrtf(d2);   // reference zero-mask semantics
    }
}

// ---------------- per-anchor reduction (N^3 mask collapsed: labels==arange) ----------------
__global__ __launch_bounds__(256)
void taw_reduce_kernel(const float* __restrict__ dist,
                       float* __restrict__ psum, float* __restrict__ pcnt, int N) {
    __shared__ float ssum[256];
    __shared__ float scnt[256];
    const int i = blockIdx.x;
    const float diag = dist[(size_t)i * N + i];
    float s = 0.0f, c = 0.0f;
    for (int n = threadIdx.x; n < N; n += 256) {
        if (n == i) continue;
        const float t = diag - dist[(size_t)i * N + n];
        if (t > 0.0f) s += t;        // relu(masked triplet)
        if (t > EPSF) c += 1.0f;     // num_positive
    }
    ssum[threadIdx.x] = s;
    scnt[threadIdx.x] = c;
    __syncthreads();
    for (int off = 128; off > 0; off >>= 1) {
        if (threadIdx.x < off) {
            ssum[threadIdx.x] += ssum[threadIdx.x + off];
            scnt[threadIdx.x] += scnt[threadIdx.x + off];
        }
        __syncthreads();
    }
    if (threadIdx.x == 0) { psum[i] = ssum[0]; pcnt[i] = scnt[0]; }
}

// ---------------- deterministic fixed-order final reduction ----------------
__global__ __launch_bounds__(256)
void taw_finalize_kernel(const float* __restrict__ psum, const float* __restrict__ pcnt,
                         float* __restrict__ out, int N) {
    __shared__ float ssum[256];
    __shared__ float scnt[256];
    float s = 0.0f, c = 0.0f;
    for (int i = threadIdx.x; i < N; i += 256) { s += psum[i]; c += pcnt[i]; }
    ssum[threadIdx.x] = s;
    scnt[threadIdx.x] = c;
    __syncthreads();
    for (int off = 128; off > 0; off >>= 1) {
        if (threadIdx.x < off) {
            ssum[threadIdx.x] += ssum[threadIdx.x + off];
            scnt[threadIdx.x] += scnt[threadIdx.x + off];
        }
        __syncthreads();
    }
    if (threadIdx.x == 0) out[0] = ssum[0] / (scnt[0] + EPSF);
}

extern "C" void kernel_launch(void* const* d_in, const int* in_sizes, int n_in,
                              void* d_out, int out_size, void* d_ws, size_t ws_size,
                              hipStream_t stream) {
    const float* a = (const float*)d_in[0];
    const float* p = (const float*)d_in[1];
    const int D = D_DIM;
    const int N = in_sizes[0] / D;              // 512 for the reference shapes

    float* dotdist = (float*)d_ws;              // N*N floats (dot, then dist in place)
    float* sqa     = dotdist + (size_t)N * N;   // N floats
    float* sqp     = sqa + N;                   // N floats
    float* psum    = sqp + N;                   // N floats
    float* pcnt    = psum + N;                  // N floats

    const int total = N * N;
    taw_init_kernel<<<(total + 255) / 256, 256, 0, stream>>>(dotdist, total);
    taw_norms_kernel<<<2 * N, 256, 0, stream>>>(a, p, sqa, sqp, N, D);
    dim3 grid(N / 64, N / 64, 2);               // 2-way K split
    taw_gemm_kernel<<<grid, 128, 0, stream>>>(a, p, dotdist, N, D);
    taw_dist_kernel<<<(total + 255) / 256, 256, 0, stream>>>(dotdist, sqa, sqp, N);
    taw_reduce_kernel<<<N, 256, 0, stream>>>(dotdist, psum, pcnt, N);
    taw_finalize_kernel<<<1, 256, 0, stream>>>(psum, pcnt, (float*)d_out, N);
}